// GCNConv_52046413693115
// MI455X (gfx1250) — compile-verified
//
#include <hip/hip_runtime.h>
#include <math.h>

typedef __attribute__((ext_vector_type(2))) float v2f;
typedef __attribute__((ext_vector_type(8))) float v8f;

#define GCN_EPS 1e-8f

// ---------------------------------------------------------------- zero scratch
__global__ void gcn_zero_kernel(float* __restrict__ p, long long n) {
    long long i = (long long)blockIdx.x * blockDim.x + threadIdx.x;
    if (i < n) p[i] = 0.0f;
}

// ---------------------------------------------------------------- in-degree
__global__ void gcn_degree_kernel(const int* __restrict__ tgt,
                                  float* __restrict__ deg, int E) {
    int e = blockIdx.x * blockDim.x + threadIdx.x;
    if (e < E) atomicAdd(&deg[tgt[e]], 1.0f);  // global_atomic_add_f32, L2-resident
}

// ---------------------------------------------------------------- normalized scatter-add
// thread = (edge, float4-of-features); 16 threads per edge. The float4 gather
// lowers to global_load_b128; src/tgt/deg are loaded once per 4 features
// instead of once per feature. The 64 float atomics per edge are the hard
// L2-throughput floor and are unchanged.
__global__ void gcn_scatter_kernel(const float4* __restrict__ x4,
                                   const int* __restrict__ src,
                                   const int* __restrict__ tgt,
                                   const float* __restrict__ deg,
                                   float* __restrict__ agg, int E) {
    long long gid = (long long)blockIdx.x * blockDim.x + threadIdx.x;
    int e  = (int)(gid >> 4);
    int f4 = (int)(gid & 15);          // 16 float4 chunks = 64 features
    if (e < E) {
        int s = src[e];
        int t = tgt[e];
        float norm = 1.0f / (sqrtf(deg[t] * deg[s]) + GCN_EPS);
        float4 v = x4[(long long)s * 16 + f4];     // global_load_b128
        float* dst = agg + (long long)t * 64 + f4 * 4;
        atomicAdd(dst + 0, v.x * norm);
        atomicAdd(dst + 1, v.y * norm);
        atomicAdd(dst + 2, v.z * norm);
        atomicAdd(dst + 3, v.w * norm);
    }
}

// ---------------------------------------------------------------- fp32 WMMA GEMM
// out[N,64] = agg[N,64] @ W^T[64,64] + deg[n]*b
// One wave32 computes a 16x16 output tile via 16 chained V_WMMA_F32_16X16X4_F32.
// Block = 256 threads = 8 waves = 2 row-tiles x 4 col-tiles (32 rows x 64 cols).
__global__ void gcn_gemm_wmma_kernel(const float* __restrict__ agg,
                                     const float* __restrict__ W,
                                     const float* __restrict__ bias,
                                     const float* __restrict__ deg,
                                     float* __restrict__ out, int N) {
    // Stage W transposed in LDS: Wt[d*64 + o] = W[o*64 + d]
    // -> B-operand reads (consecutive lanes = consecutive cols) are conflict-free.
    __shared__ float Wt[64 * 64];
    for (int i = threadIdx.x; i < 64 * 64; i += blockDim.x) {
        int o = i >> 6;
        int d = i & 63;
        Wt[d * 64 + o] = W[i];
    }
    __syncthreads();

    const int wave = threadIdx.x >> 5;          // 0..7
    const int lane = threadIdx.x & 31;
    const int rowTile = blockIdx.x * 2 + (wave >> 2);
    const int colTile = wave & 3;
    const int row0 = rowTile * 16;
    if (row0 + 15 >= N) return;                 // wave-uniform: EXEC stays all-ones

    const int m = lane & 15;                    // M index for A, N index for B/C/D
    const int h = lane >> 4;                    // half-wave selector
    const int col = colTile * 16 + m;
    const float* __restrict__ arow = agg + (long long)(row0 + m) * 64;

    v8f c = {};
    #pragma unroll
    for (int kk = 0; kk < 16; ++kk) {
        // A 16x4 f32 layout: lanes 0-15 hold K=k0,k0+1; lanes 16-31 hold K=k0+2,k0+3
        const int k0 = kk * 4 + 2 * h;
        v2f a;
        a.x = arow[k0 + 0];
        a.y = arow[k0 + 1];
        // B 4x16 layout: row K striped across lanes; half-waves split the K-range
        v2f b;
        b.x = Wt[(k0 + 0) * 64 + col];
        b.y = Wt[(k0 + 1) * 64 + col];
        c = __builtin_amdgcn_wmma_f32_16x16x4_f32(
                /*neg_a=*/false, a, /*neg_b=*/false, b,
                /*c_mod=*/(short)0, c, /*reuse_a=*/false, /*reuse_b=*/false);
    }

    // C/D layout: VGPR v -> (M = v + 8*h, N = m). Reference adds bias per edge,
    // which after aggregation becomes deg[n] * b.
    const float bc = bias[col];
    #pragma unroll
    for (int v = 0; v < 8; ++v) {
        int r = row0 + v + 8 * h;
        out[(long long)r * 64 + col] = c[v] + deg[r] * bc;
    }
}

// ---------------------------------------------------------------- launcher
extern "C" void kernel_launch(void* const* d_in, const int* in_sizes, int n_in,
                              void* d_out, int out_size, void* d_ws, size_t ws_size,
                              hipStream_t stream) {
    const float* x   = (const float*)d_in[0];   // [N,64]
    const int*   ei  = (const int*)d_in[1];     // [2,E]
    const float* W   = (const float*)d_in[2];   // [64,64]
    const float* b   = (const float*)d_in[3];   // [64]
    float* out = (float*)d_out;

    const int N = in_sizes[0] / 64;
    const int E = in_sizes[1] / 2;
    const int* src = ei;
    const int* tgt = ei + E;

    // scratch layout: deg [N floats] | agg [N*64 floats]
    float* deg = (float*)d_ws;
    float* agg = deg + N;

    // 1) zero deg + agg (harness does not re-zero between replays)
    long long zn = (long long)N * 65;
    gcn_zero_kernel<<<(int)((zn + 255) / 256), 256, 0, stream>>>(deg, zn);

    // 2) in-degree
    gcn_degree_kernel<<<(E + 255) / 256, 256, 0, stream>>>(tgt, deg, E);

    // 3) normalized gather/scatter into agg (16 threads per edge, float4 chunks)
    long long sthreads = (long long)E * 16;
    gcn_scatter_kernel<<<(int)((sthreads + 255) / 256), 256, 0, stream>>>(
        (const float4*)x, src, tgt, deg, agg, E);

    // 4) fp32 WMMA GEMM + degree-scaled bias
    int rowTiles = (N + 15) / 16;
    int blocks = (rowTiles + 1) / 2;
    gcn_gemm_wmma_kernel<<<blocks, 256, 0, stream>>>(agg, W, b, deg, out, N);
}